// GCNGraphClassifier_35656818491728
// MI455X (gfx1250) — compile-verified
//
#include <hip/hip_runtime.h>

typedef float v2f __attribute__((ext_vector_type(2)));
typedef float v8f __attribute__((ext_vector_type(8)));

#define NGRAPHS 1024
#define THREADS 256

// ---- prep: deg=1 (self-loop), pack x->[N,4] (K pad), transpose W1->[16,4], W2->[32,16],
//      zero pool accumulators ----
__global__ void k_prep(const float* __restrict__ x, const float* __restrict__ W1,
                       const float* __restrict__ W2,
                       float* __restrict__ deg, float* __restrict__ x4,
                       float* __restrict__ W1T, float* __restrict__ W2T,
                       float* __restrict__ sums, float* __restrict__ cnt, int N) {
  int i = blockIdx.x * blockDim.x + threadIdx.x;
  if (i < N) {
    deg[i] = 1.0f;                          // self-loop contributes 1 to every node
    x4[i * 4 + 0] = x[i * 3 + 0];
    x4[i * 4 + 1] = x[i * 3 + 1];
    x4[i * 4 + 2] = x[i * 3 + 2];
    x4[i * 4 + 3] = 0.0f;                   // K=3 -> 4 zero pad
  }
  if (i < 64) {                             // W1T[n*4+k] = W1[k][n], k==3 -> 0
    int n = i >> 2, k = i & 3;
    W1T[i] = (k < 3) ? W1[k * 16 + n] : 0.0f;
  }
  if (i < 512) {                            // W2T[c*16+k] = W2[k][c]
    int c = i >> 4, k = i & 15;
    W2T[i] = W2[k * 32 + c];
  }
  if (i < NGRAPHS * 32) sums[i] = 0.0f;
  if (i < NGRAPHS) cnt[i] = 0.0f;
}

// ---- degree scatter over edges ----
__global__ void k_deg(const long long* __restrict__ dst, float* __restrict__ deg, int E) {
  int i = blockIdx.x * blockDim.x + threadIdx.x;
  if (i < E) atomicAdd(&deg[(int)dst[i]], 1.0f);
}

// ---- deg -> d^{-1/2} in place ----
__global__ void k_dinv(float* __restrict__ deg, int N) {
  int i = blockIdx.x * blockDim.x + threadIdx.x;
  if (i < N) deg[i] = rsqrtf(deg[i]);       // deg >= 1 always (self-loop)
}

// ---- layer 1 GEMM: hs1 = dinv * (x @ W1), agg1 seeded with self-loop term ----
// One wave = one 16-node tile = one V_WMMA_F32_16X16X4_F32. All loads branch-free v2f.
__global__ void k_gemm1(const float* __restrict__ x4, const float* __restrict__ W1T,
                        const float* __restrict__ dinv,
                        float* __restrict__ hs1, float* __restrict__ agg1, int N) {
  const int lane = threadIdx.x & 31;
  const int wave = threadIdx.x >> 5;
  const int tile = blockIdx.x * (blockDim.x >> 5) + wave;
  const int base = tile * 16;
  if (base >= N) return;                    // wave-uniform
  const int half = lane >> 4;               // K split: 0 -> K0,K1 ; 1 -> K2,K3
  const int m    = lane & 15;
  int arow = base + m; if (arow >= N) arow = N - 1;

  v2f a = *(const v2f*)(x4  + (size_t)arow * 4 + half * 2);
  v2f b = *(const v2f*)(W1T + m * 4 + half * 2);
  v8f c = {};
  c = __builtin_amdgcn_wmma_f32_16x16x4_f32(false, a, false, b, (short)0, c, false, false);

  if (base + 16 <= N) {                     // hot path: unconditional stores
#pragma unroll
    for (int r = 0; r < 8; ++r) {
      int grow = base + half * 8 + r;
      float v = dinv[grow] * c[r];
      hs1[grow * 16 + m]  = v;
      agg1[grow * 16 + m] = v;
    }
  } else {
#pragma unroll
    for (int r = 0; r < 8; ++r) {
      int grow = base + half * 8 + r;
      if (grow < N) {
        float v = dinv[grow] * c[r];
        hs1[grow * 16 + m]  = v;
        agg1[grow * 16 + m] = v;
      }
    }
  }
}

// ---- edge scatter, layer 1: thread = (edge, 4-feature chunk) ----
__global__ void k_scat1(const long long* __restrict__ src, const long long* __restrict__ dst,
                        const float* __restrict__ hs1, float* __restrict__ agg1, int E) {
  long long id = (long long)blockIdx.x * blockDim.x + threadIdx.x;
  if (id >= (long long)E * 4) return;
  int e = (int)(id >> 2), q = (int)(id & 3);
  int s = (int)src[e], d = (int)dst[e];
  float4 v = *(const float4*)(hs1 + (size_t)s * 16 + q * 4);
  float* p = agg1 + (size_t)d * 16 + q * 4;
  atomicAdd(p + 0, v.x);
  atomicAdd(p + 1, v.y);
  atomicAdd(p + 2, v.z);
  atomicAdd(p + 3, v.w);
}

// ---- finalize layer 1: h1 = relu(dinv*agg1 + b1), float4 ----
__global__ void k_fin1(const float* __restrict__ agg1, const float* __restrict__ dinv,
                       const float* __restrict__ b1, float* __restrict__ h1f, int N) {
  long long id = (long long)blockIdx.x * blockDim.x + threadIdx.x;
  if (id >= (long long)N * 4) return;
  int v = (int)(id >> 2), q = (int)(id & 3);
  float dv = dinv[v];
  float4 a  = *(const float4*)(agg1 + id * 4);
  float4 bb = *(const float4*)(b1 + q * 4);
  float4 o;
  o.x = fmaxf(dv * a.x + bb.x, 0.0f);
  o.y = fmaxf(dv * a.y + bb.y, 0.0f);
  o.z = fmaxf(dv * a.z + bb.z, 0.0f);
  o.w = fmaxf(dv * a.w + bb.w, 0.0f);
  *(float4*)(h1f + id * 4) = o;
}

// ---- layer 2 GEMM: hs2 = dinv * (h1 @ W2), agg2 seeded. 8 chained WMMAs/wave ----
__global__ void k_gemm2(const float* __restrict__ h1f, const float* __restrict__ W2T,
                        const float* __restrict__ dinv,
                        float* __restrict__ hs2, float* __restrict__ agg2, int N) {
  const int lane = threadIdx.x & 31;
  const int wave = threadIdx.x >> 5;
  const int tile = blockIdx.x * (blockDim.x >> 5) + wave;
  const int base = tile * 16;
  if (base >= N) return;
  const int half = lane >> 4;
  const int m    = lane & 15;
  int arow = base + m; if (arow >= N) arow = N - 1;

  v8f c0 = {}, c1 = {};
#pragma unroll
  for (int kk = 0; kk < 4; ++kk) {
    const int k0 = kk * 4 + half * 2;       // this lane-half's two K values
    v2f a  = *(const v2f*)(h1f + (size_t)arow * 16 + k0);
    v2f b0 = *(const v2f*)(W2T + m * 16 + k0);
    v2f b1 = *(const v2f*)(W2T + (m + 16) * 16 + k0);
    c0 = __builtin_amdgcn_wmma_f32_16x16x4_f32(false, a, false, b0, (short)0, c0, false, false);
    c1 = __builtin_amdgcn_wmma_f32_16x16x4_f32(false, a, false, b1, (short)0, c1, false, false);
  }

  if (base + 16 <= N) {
#pragma unroll
    for (int r = 0; r < 8; ++r) {
      int grow = base + half * 8 + r;
      float dv = dinv[grow];
      float v0 = dv * c0[r], v1 = dv * c1[r];
      hs2[(size_t)grow * 32 + m]       = v0;  agg2[(size_t)grow * 32 + m]       = v0;
      hs2[(size_t)grow * 32 + 16 + m]  = v1;  agg2[(size_t)grow * 32 + 16 + m]  = v1;
    }
  } else {
#pragma unroll
    for (int r = 0; r < 8; ++r) {
      int grow = base + half * 8 + r;
      if (grow < N) {
        float dv = dinv[grow];
        float v0 = dv * c0[r], v1 = dv * c1[r];
        hs2[(size_t)grow * 32 + m]       = v0;  agg2[(size_t)grow * 32 + m]       = v0;
        hs2[(size_t)grow * 32 + 16 + m]  = v1;  agg2[(size_t)grow * 32 + 16 + m]  = v1;
      }
    }
  }
}

// ---- edge scatter, layer 2: thread = (edge, 4-feature chunk), 8 chunks/edge ----
__global__ void k_scat2(const long long* __restrict__ src, const long long* __restrict__ dst,
                        const float* __restrict__ hs2, float* __restrict__ agg2, int E) {
  long long id = (long long)blockIdx.x * blockDim.x + threadIdx.x;
  if (id >= (long long)E * 8) return;
  int e = (int)(id >> 3), q = (int)(id & 7);
  int s = (int)src[e], d = (int)dst[e];
  float4 v = *(const float4*)(hs2 + (size_t)s * 32 + q * 4);
  float* p = agg2 + (size_t)d * 32 + q * 4;
  atomicAdd(p + 0, v.x);
  atomicAdd(p + 1, v.y);
  atomicAdd(p + 2, v.z);
  atomicAdd(p + 3, v.w);
}

// ---- finalize layer 2 + mean-pool accumulation (float4 per thread) ----
__global__ void k_fin2_pool(const float* __restrict__ agg2, const float* __restrict__ dinv,
                            const float* __restrict__ b2, const long long* __restrict__ batch,
                            float* __restrict__ sums, float* __restrict__ cnt, int N) {
  long long id = (long long)blockIdx.x * blockDim.x + threadIdx.x;
  if (id >= (long long)N * 8) return;
  int v = (int)(id >> 3), q = (int)(id & 7);
  float dv = dinv[v];
  float4 a  = *(const float4*)(agg2 + id * 4);
  float4 bb = *(const float4*)(b2 + q * 4);
  float4 t;
  t.x = fmaxf(dv * a.x + bb.x, 0.0f);
  t.y = fmaxf(dv * a.y + bb.y, 0.0f);
  t.z = fmaxf(dv * a.z + bb.z, 0.0f);
  t.w = fmaxf(dv * a.w + bb.w, 0.0f);
  int g = (int)batch[v];
  float* p = sums + (size_t)g * 32 + q * 4;
  atomicAdd(p + 0, t.x);
  atomicAdd(p + 1, t.y);
  atomicAdd(p + 2, t.z);
  atomicAdd(p + 3, t.w);
  if (q == 0) atomicAdd(&cnt[g], 1.0f);
}

// ---- head: out = (sums/max(cnt,1)) @ Wl + bl ----
__global__ void k_head(const float* __restrict__ sums, const float* __restrict__ cnt,
                       const float* __restrict__ Wl, const float* __restrict__ bl,
                       float* __restrict__ out) {
  int id = blockIdx.x * blockDim.x + threadIdx.x;
  if (id >= NGRAPHS * 3) return;
  int g = id / 3, j = id % 3;
  float c = cnt[g]; c = c > 1.0f ? c : 1.0f;
  float inv = 1.0f / c;
  float acc = bl[j];
#pragma unroll
  for (int f = 0; f < 32; ++f) acc += (sums[g * 32 + f] * inv) * Wl[f * 3 + j];
  out[id] = acc;
}

static inline size_t align16f(size_t n) { return (n + 15) & ~(size_t)15; }

extern "C" void kernel_launch(void* const* d_in, const int* in_sizes, int n_in,
                              void* d_out, int out_size, void* d_ws, size_t ws_size,
                              hipStream_t stream) {
  const float*     x    = (const float*)d_in[0];
  const long long* eidx = (const long long*)d_in[1];
  const long long* bat  = (const long long*)d_in[2];
  const float*     W1   = (const float*)d_in[3];
  const float*     b1   = (const float*)d_in[4];
  const float*     W2   = (const float*)d_in[5];
  const float*     b2   = (const float*)d_in[6];
  const float*     Wl   = (const float*)d_in[7];
  const float*     bl   = (const float*)d_in[8];
  float*           out  = (float*)d_out;

  const int N = in_sizes[0] / 3;
  const int E = in_sizes[1] / 2;
  const long long* src = eidx;
  const long long* dst = eidx + E;

  // workspace layout (floats), regions 64B-aligned
  float* ws   = (float*)d_ws;
  size_t off  = 0;
  float* dinv = ws + off; off += align16f((size_t)N);           // deg -> dinv
  float* x4   = ws + off; off += align16f((size_t)N * 4);       // padded x
  float* buf1 = ws + off; off += align16f((size_t)N * 16);      // hs1, then h1f
  float* agg1 = ws + off; off += align16f((size_t)N * 16);
  float* hs2  = ws + off; off += align16f((size_t)N * 32);
  float* agg2 = ws + off; off += align16f((size_t)N * 32);
  float* sums = ws + off; off += align16f((size_t)NGRAPHS * 32);
  float* cnt  = ws + off; off += align16f((size_t)NGRAPHS);
  float* W1T  = ws + off; off += align16f(64);
  float* W2T  = ws + off; off += align16f(512);

  const int B = THREADS;
  const int initMax = (N > NGRAPHS * 32) ? N : NGRAPHS * 32;
  const int tiles = (N + 15) / 16;
  const int wavesPerBlk = B / 32;
  const unsigned gemmBlocks = (tiles + wavesPerBlk - 1) / wavesPerBlk;

  k_prep<<<(initMax + B - 1) / B, B, 0, stream>>>(x, W1, W2, dinv, x4, W1T, W2T, sums, cnt, N);
  k_deg <<<(E + B - 1) / B, B, 0, stream>>>(dst, dinv, E);
  k_dinv<<<(N + B - 1) / B, B, 0, stream>>>(dinv, N);

  k_gemm1<<<gemmBlocks, B, 0, stream>>>(x4, W1T, dinv, buf1, agg1, N);

  long long w1 = (long long)E * 4;
  k_scat1<<<(unsigned)((w1 + B - 1) / B), B, 0, stream>>>(src, dst, buf1, agg1, E);

  long long n1 = (long long)N * 4;
  k_fin1<<<(unsigned)((n1 + B - 1) / B), B, 0, stream>>>(agg1, dinv, b1, buf1, N); // buf1 := h1f

  k_gemm2<<<gemmBlocks, B, 0, stream>>>(buf1, W2T, dinv, hs2, agg2, N);

  long long w2 = (long long)E * 8;
  k_scat2<<<(unsigned)((w2 + B - 1) / B), B, 0, stream>>>(src, dst, hs2, agg2, E);

  long long n2 = (long long)N * 8;
  k_fin2_pool<<<(unsigned)((n2 + B - 1) / B), B, 0, stream>>>(agg2, dinv, b2, bat, sums, cnt, N);

  k_head<<<(NGRAPHS * 3 + B - 1) / B, B, 0, stream>>>(sums, cnt, Wl, bl, out);
}